// Net_22093311770979
// MI455X (gfx1250) — compile-verified
//
#include <hip/hip_runtime.h>

typedef __attribute__((ext_vector_type(2))) float v2f;
typedef __attribute__((ext_vector_type(8))) float v8f;

#define NN   50000
#define NE   800000
#define FIN  9
#define H1D  512
#define H2D  128
#define NC   2

// ================= CSR-by-dst construction =================
__global__ void k_init_cnt(int* cnt) {
    int i = blockIdx.x * blockDim.x + threadIdx.x;
    if (i < NN) cnt[i] = 0;
}
__global__ void k_count(const int* __restrict__ ei, int* cnt) {
    int e = blockIdx.x * blockDim.x + threadIdx.x;
    if (e < NE) atomicAdd(&cnt[ei[NE + e]], 1);            // dst = ei[1][e]
}
__global__ void k_dinv(const int* __restrict__ cnt, float* __restrict__ dinv) {
    int i = blockIdx.x * blockDim.x + threadIdx.x;
    if (i < NN) dinv[i] = rsqrtf((float)cnt[i] + 1.0f);    // +1 self-loop
}
// single-workgroup exclusive scan over 50k counts -> rowptr (+ cursor copy)
__global__ __launch_bounds__(1024) void k_scan(const int* __restrict__ cnt,
                                               int* __restrict__ rowptr,
                                               int* __restrict__ cursor) {
    __shared__ int sums[1024];
    const int t  = threadIdx.x;
    const int CH = (NN + 1023) / 1024;                     // 49 nodes/thread
    int lo = t * CH;
    int hi = lo + CH; if (hi > NN) hi = NN; if (lo > NN) lo = NN;
    int s = 0;
    for (int i = lo; i < hi; ++i) s += cnt[i];
    sums[t] = s;
    __syncthreads();
    for (int off = 1; off < 1024; off <<= 1) {             // inclusive scan
        int v = sums[t];
        int u = (t >= off) ? sums[t - off] : 0;
        __syncthreads();
        sums[t] = v + u;
        __syncthreads();
    }
    int base = (t == 0) ? 0 : sums[t - 1];
    for (int i = lo; i < hi; ++i) {
        rowptr[i] = base;
        cursor[i] = base;
        base += cnt[i];
    }
    if (t == 1023) rowptr[NN] = base;                      // == NE
}
__global__ void k_fill(const int* __restrict__ ei, const float* __restrict__ dinv,
                       int* cursor, int* __restrict__ srcs, float* __restrict__ wts) {
    int e = blockIdx.x * blockDim.x + threadIdx.x;
    if (e >= NE) return;
    int s = ei[e], d = ei[NE + e];
    int slot = atomicAdd(&cursor[d], 1);
    srcs[slot] = s;
    wts[slot]  = dinv[s] * dinv[d];
}

// ================= layer 1: gather X (9-wide), then GEMM =================
__global__ void k_agg1(const float* __restrict__ x, const float* __restrict__ dinv,
                       const int* __restrict__ rowptr, const int* __restrict__ srcs,
                       const float* __restrict__ wts, float* __restrict__ xa) {
    int i = blockIdx.x * blockDim.x + threadIdx.x;
    if (i >= NN * FIN) return;
    int n = i / FIN, f = i - n * FIN;
    float d = dinv[n];
    float acc = x[i] * d * d;                              // self-loop
    int beg = rowptr[n], end = rowptr[n + 1];
    for (int j = beg; j < end; ++j)
        acc = fmaf(x[(size_t)srcs[j] * FIN + f], wts[j], acc);
    xa[i] = acc;
}
__global__ void k_gemm1_relu(const float* __restrict__ xa, const float* __restrict__ w1,
                             const float* __restrict__ b1, float* __restrict__ h1) {
    int i = blockIdx.x * blockDim.x + threadIdx.x;
    if (i >= NN * H1D) return;
    int row = i / H1D, col = i - row * H1D;
    const float* xr = xa + (size_t)row * FIN;
    float acc = b1[col];
    #pragma unroll
    for (int f = 0; f < FIN; ++f) acc = fmaf(xr[f], w1[f * H1D + col], acc);
    h1[i] = fmaxf(acc, 0.0f);
}

// ====== layer 2 GEMM: H1(Nx512) @ W2(512x128) via V_WMMA_F32_16X16X4_F32 ======
#define LDS_STRIDE 516   // 516 % 64 = 4 -> 16 lanes hit 16 distinct banks
__global__ __launch_bounds__(256) void k_gemm2_wmma(const float* __restrict__ h1,
                                                    const float* __restrict__ w2,
                                                    float* __restrict__ p2) {
    __shared__ float lds[16 * LDS_STRIDE];
    const int row0 = blockIdx.x * 16;
    const int tid = threadIdx.x;

    // 16 consecutive rows of H1 = one contiguous 32KB span -> coalesced panel load
    const float4* src = (const float4*)(h1 + (size_t)row0 * H1D);
    for (int i = tid; i < 16 * (H1D / 4); i += 256) {
        float4 v = src[i];
        int r = i >> 7;
        int c = (i & 127) << 2;
        float* dp = &lds[r * LDS_STRIDE + c];
        dp[0] = v.x; dp[1] = v.y; dp[2] = v.z; dp[3] = v.w;
    }
    __syncthreads();

    const int wave = tid >> 5;          // 0..7 -> column tile
    const int lane = tid & 31;
    const int m    = lane & 15;
    const int koff = (lane >> 4) << 1;  // lanes 16-31 hold K+2,K+3
    const int col  = (wave << 4) + m;

    v8f c = {};
    for (int k = 0; k < H1D; k += 4) {
        v2f a, b;
        const float* ap = &lds[m * LDS_STRIDE + k + koff];
        a.x = ap[0];
        a.y = ap[1];
        const float* bp = &w2[(size_t)(k + koff) * H2D + col];
        b.x = bp[0];
        b.y = bp[H2D];
        c = __builtin_amdgcn_wmma_f32_16x16x4_f32(false, a, false, b,
                                                  (short)0, c, false, false);
    }
    const int rsel = (lane >> 4) << 3;
    #pragma unroll
    for (int vv = 0; vv < 8; ++vv) {
        int r = row0 + vv + rsel;
        p2[(size_t)r * H2D + col] = c[vv];
    }
}

// ====== layer 2 aggregation: one wave per node, lane = float4 chunk (no atomics) ======
__global__ __launch_bounds__(256) void k_agg2(const float* __restrict__ p2,
                                              const float* __restrict__ dinv,
                                              const int* __restrict__ rowptr,
                                              const int* __restrict__ srcs,
                                              const float* __restrict__ wts,
                                              const float* __restrict__ b2,
                                              float* __restrict__ h2) {
    int wv = (blockIdx.x * blockDim.x + threadIdx.x) >> 5; // node
    if (wv >= NN) return;
    int lane = threadIdx.x & 31;
    int q = lane << 2;                                     // 32 lanes x float4 = 128
    float d = dinv[wv];
    float d2 = d * d;
    float4 acc = *(const float4*)(p2 + (size_t)wv * H2D + q);
    acc.x *= d2; acc.y *= d2; acc.z *= d2; acc.w *= d2;    // self-loop
    int beg = rowptr[wv], end = rowptr[wv + 1];
    for (int j = beg; j < end; ++j) {                      // wave-uniform metadata
        int s   = srcs[j];
        float w = wts[j];
        float4 v = *(const float4*)(p2 + (size_t)s * H2D + q); // coalesced 512B row
        acc.x = fmaf(v.x, w, acc.x);
        acc.y = fmaf(v.y, w, acc.y);
        acc.z = fmaf(v.z, w, acc.z);
        acc.w = fmaf(v.w, w, acc.w);
    }
    const float4 bb = *(const float4*)(b2 + q);
    float4 o;
    o.x = fmaxf(acc.x + bb.x, 0.0f);
    o.y = fmaxf(acc.y + bb.y, 0.0f);
    o.z = fmaxf(acc.z + bb.z, 0.0f);
    o.w = fmaxf(acc.w + bb.w, 0.0f);
    *(float4*)(h2 + (size_t)wv * H2D + q) = o;
}

// ================= layer 3: GEMM (128->2), gather, log_softmax =================
__global__ void k_gemm3(const float* __restrict__ h2, const float* __restrict__ w3,
                        float* __restrict__ p3) {
    int i = blockIdx.x * blockDim.x + threadIdx.x;
    if (i >= NN) return;
    const float4* hr = (const float4*)(h2 + (size_t)i * H2D);
    float t0 = 0.0f, t1 = 0.0f;
    #pragma unroll 4
    for (int k = 0; k < H2D / 4; ++k) {
        float4 v = hr[k];
        const float* wp = w3 + k * 8;
        t0 = fmaf(v.x, wp[0], t0); t1 = fmaf(v.x, wp[1], t1);
        t0 = fmaf(v.y, wp[2], t0); t1 = fmaf(v.y, wp[3], t1);
        t0 = fmaf(v.z, wp[4], t0); t1 = fmaf(v.z, wp[5], t1);
        t0 = fmaf(v.w, wp[6], t0); t1 = fmaf(v.w, wp[7], t1);
    }
    p3[i * 2 + 0] = t0;
    p3[i * 2 + 1] = t1;
}
__global__ void k_final(const float* __restrict__ p3, const float* __restrict__ dinv,
                        const int* __restrict__ rowptr, const int* __restrict__ srcs,
                        const float* __restrict__ wts, const float* __restrict__ b3,
                        float* __restrict__ out) {
    int i = blockIdx.x * blockDim.x + threadIdx.x;
    if (i >= NN) return;
    float d = dinv[i], d2 = d * d;
    float a0 = p3[i * 2 + 0] * d2;
    float a1 = p3[i * 2 + 1] * d2;
    int beg = rowptr[i], end = rowptr[i + 1];
    for (int j = beg; j < end; ++j) {
        int s   = srcs[j];
        float w = wts[j];
        a0 = fmaf(p3[s * 2 + 0], w, a0);
        a1 = fmaf(p3[s * 2 + 1], w, a1);
    }
    a0 += b3[0];
    a1 += b3[1];
    float m = fmaxf(a0, a1);
    float lse = m + logf(expf(a0 - m) + expf(a1 - m));
    out[i * 2 + 0] = a0 - lse;
    out[i * 2 + 1] = a1 - lse;
}

// ================= host launcher =================
extern "C" void kernel_launch(void* const* d_in, const int* in_sizes, int n_in,
                              void* d_out, int out_size, void* d_ws, size_t ws_size,
                              hipStream_t stream) {
    const float* x  = (const float*)d_in[0];
    const float* w1 = (const float*)d_in[1];
    const float* b1 = (const float*)d_in[2];
    const float* w2 = (const float*)d_in[3];
    const float* b2 = (const float*)d_in[4];
    const float* w3 = (const float*)d_in[5];
    const float* b3 = (const float*)d_in[6];
    const int*   ei = (const int*)d_in[7];
    float* out = (float*)d_out;

    size_t off = 0;
    auto carve = [&](size_t bytes) -> void* {
        void* p = (char*)d_ws + off;
        off = (off + bytes + 255) & ~(size_t)255;
        return p;
    };
    int*   cnt    = (int*)  carve((size_t)NN * 4);
    int*   rowptr = (int*)  carve((size_t)(NN + 1) * 4);
    int*   cursor = (int*)  carve((size_t)NN * 4);
    int*   srcs   = (int*)  carve((size_t)NE * 4);
    float* wts    = (float*)carve((size_t)NE * 4);
    float* dinv   = (float*)carve((size_t)NN * 4);
    float* xa     = (float*)carve((size_t)NN * FIN * 4);
    float* h1     = (float*)carve((size_t)NN * H1D * 4);
    float* p2     = (float*)carve((size_t)NN * H2D * 4);
    float* h2     = (float*)carve((size_t)NN * H2D * 4);
    float* p3     = (float*)carve((size_t)NN * NC * 4);

    const int B = 256;
    auto g = [&](long long n) { return (int)((n + B - 1) / B); };

    // CSR build (one graph, reused by all 3 layers)
    k_init_cnt<<<g(NN), B, 0, stream>>>(cnt);
    k_count   <<<g(NE), B, 0, stream>>>(ei, cnt);
    k_dinv    <<<g(NN), B, 0, stream>>>(cnt, dinv);
    k_scan    <<<1, 1024, 0, stream>>>(cnt, rowptr, cursor);
    k_fill    <<<g(NE), B, 0, stream>>>(ei, dinv, cursor, srcs, wts);

    // layer 1: aggregate 9-wide first (A@X)@W1 == A@(X@W1)
    k_agg1      <<<g((long long)NN * FIN), B, 0, stream>>>(x, dinv, rowptr, srcs, wts, xa);
    k_gemm1_relu<<<g((long long)NN * H1D), B, 0, stream>>>(xa, w1, b1, h1);

    // layer 2: WMMA GEMM then 128-wide gather (+bias+relu fused)
    k_gemm2_wmma<<<NN / 16, B, 0, stream>>>(h1, w2, p2);
    k_agg2      <<<(NN * 32 + B - 1) / B, B, 0, stream>>>(p2, dinv, rowptr, srcs, wts, b2, h2);

    // layer 3: 128->2 GEMM, gather + bias + log_softmax fused
    k_gemm3<<<g(NN), B, 0, stream>>>(h2, w3, p3);
    k_final<<<g(NN), B, 0, stream>>>(p3, dinv, rowptr, srcs, wts, b3, out);
}